// minibatch_discriminator_72430328480859
// MI455X (gfx1250) — compile-verified
//
#include <hip/hip_runtime.h>
#include <hip/hip_bf16.h>

typedef float v2f __attribute__((ext_vector_type(2)));
typedef float v8f __attribute__((ext_vector_type(8)));

#define N_SAMP 256
#define FEAT   8192
#define BDIM   128
#define CDIM   16
#define NCOL   (BDIM * CDIM)     // 2048
#define OUTW   (FEAT + BDIM)     // 8320
#define KB     64
#define NCHUNK (FEAT / KB)       // 128
#define APITCH 68                // half-waves read disjoint bank sets
#define BPITCH 80                // half-waves offset by 32 banks

// Async DMA: copy 16B from global to LDS, tracked by ASYNCcnt (no VGPR data).
__device__ __forceinline__ void async_cp16(unsigned ldsOff, const float* g) {
  asm volatile("global_load_async_to_lds_b128 %0, %1, off"
               :: "v"(ldsOff), "v"(g) : "memory");
}
__device__ __forceinline__ void wait_async_le6() {
  asm volatile("s_wait_asynccnt 0x6" ::: "memory");
}
__device__ __forceinline__ void wait_async_0() {
  asm volatile("s_wait_asynccnt 0x0" ::: "memory");
}

// One K-chunk of WMMAs: preload all 32 fragments, then 16 back-to-back WMMAs.
// A 16x4 f32: lane l16 = row; v[0]=K(ka), v[1]=K(ka+1), ka = 4i + 2*half.
// B  4x16 f32: rows ka, ka+1 striped across lanes (pitch BPITCH).
__device__ __forceinline__ void chunk_wmma(const float* __restrict__ ap,
                                           const float* __restrict__ bp,
                                           v8f& acc) {
  v2f af[16], bf[16];
  #pragma unroll
  for (int i = 0; i < 16; ++i) {
    af[i].x = ap[4 * i];
    af[i].y = ap[4 * i + 1];
    bf[i].x = bp[(4 * i)     * BPITCH];
    bf[i].y = bp[(4 * i + 1) * BPITCH];
  }
  #pragma unroll
  for (int i = 0; i < 16; ++i)
    acc = __builtin_amdgcn_wmma_f32_16x16x4_f32(
        false, af[i], false, bf[i], (short)0, acc, false, false);
}

// ---------------------------------------------------------------------------
// Kernel 1: M = inp @ T via V_WMMA_F32_16X16X4_F32.
// Block = 256 threads = 8 waves; block computes a 32x64 tile of M.
// Global->LDS staging uses GLOBAL_LOAD_ASYNC_TO_LDS_B128 into a double
// buffer: the DMA for chunk c+1 streams while chunk c's WMMAs run, and no
// registers hold staged data (fixes the scratch spills of the reg pipeline).
// ---------------------------------------------------------------------------
__global__ __launch_bounds__(256) void mbd_gemm_wmma(
    const float* __restrict__ A,    // [256, 8192] row-major
    const float* __restrict__ Bm,   // [8192, 2048] row-major
    float* __restrict__ Mws)        // [128][256][16]
{
  __shared__ float As[2][32 * APITCH];
  __shared__ float Bs[2][KB * BPITCH];

  const int tid   = threadIdx.x;
  const int lane  = tid & 31;
  const int half  = lane >> 4;     // 0: lanes 0-15, 1: lanes 16-31
  const int l16   = lane & 15;
  const int wv    = tid >> 5;
  const int rtile = wv >> 2;       // 0..1
  const int ctile = wv & 3;        // 0..3

  const int rowBase = blockIdx.y * 32;   // sample rows
  const int colBase = blockIdx.x * 64;   // M columns (b*16 + c)

  // Per-thread staging geometry: A ids {tid,256+tid}; B ids {i*256+tid}.
  // r = id/16, cv = id%16; each async op moves one float4.
  int rA[2], cA[2], rB[4], cB[4];
  #pragma unroll
  for (int i = 0; i < 2; ++i) { const int id = i * 256 + tid; rA[i] = id >> 4; cA[i] = id & 15; }
  #pragma unroll
  for (int i = 0; i < 4; ++i) { const int id = i * 256 + tid; rB[i] = id >> 4; cB[i] = id & 15; }

  // LDS destination offsets (bytes) for both buffers.
  unsigned aOff[2][2], bOff[2][4];
  #pragma unroll
  for (int p = 0; p < 2; ++p) {
    #pragma unroll
    for (int i = 0; i < 2; ++i)
      aOff[p][i] = (unsigned)(unsigned long long)&As[p][rA[i] * APITCH + cA[i] * 4];
    #pragma unroll
    for (int i = 0; i < 4; ++i)
      bOff[p][i] = (unsigned)(unsigned long long)&Bs[p][rB[i] * BPITCH + cB[i] * 4];
  }

  // Per-thread global source pointers for chunk 0.
  const float* gA[2];
  const float* gB[4];
  #pragma unroll
  for (int i = 0; i < 2; ++i) gA[i] = A  + (size_t)(rowBase + rA[i]) * FEAT + cA[i] * 4;
  #pragma unroll
  for (int i = 0; i < 4; ++i) gB[i] = Bm + (size_t)rB[i] * NCOL + colBase + cB[i] * 4;

  const float* ap0 = &As[0][(rtile * 16 + l16) * APITCH + 2 * half];
  const float* bp0 = &Bs[0][2 * half * BPITCH + ctile * 16 + l16];
  const float* ap1 = &As[1][(rtile * 16 + l16) * APITCH + 2 * half];
  const float* bp1 = &Bs[1][2 * half * BPITCH + ctile * 16 + l16];

  v8f acc = {};

  // Prologue: DMA chunk 0 into buffer 0.
  #pragma unroll
  for (int i = 0; i < 2; ++i) async_cp16(aOff[0][i], gA[i]);
  #pragma unroll
  for (int i = 0; i < 4; ++i) async_cp16(bOff[0][i], gB[i]);

  for (int c = 0; c < NCHUNK; c += 2) {
    // DMA chunk c+1 into buffer 1 (c+1 <= 127 always).
    {
      const int kk = (c + 1) * KB;
      #pragma unroll
      for (int i = 0; i < 2; ++i) async_cp16(aOff[1][i], gA[i] + kk);
      #pragma unroll
      for (int i = 0; i < 4; ++i) async_cp16(bOff[1][i], gB[i] + (size_t)kk * NCOL);
    }
    wait_async_le6();      // chunk c resident (in-order completion)
    __syncthreads();       // ... across all waves
    chunk_wmma(ap0, bp0, acc);
    __syncthreads();       // buffer 0 free for refill

    if (c + 2 < NCHUNK) {
      const int kk = (c + 2) * KB;
      #pragma unroll
      for (int i = 0; i < 2; ++i) async_cp16(aOff[0][i], gA[i] + kk);
      #pragma unroll
      for (int i = 0; i < 4; ++i) async_cp16(bOff[0][i], gB[i] + (size_t)kk * NCOL);
      wait_async_le6();    // chunk c+1 resident
    } else {
      wait_async_0();
    }
    __syncthreads();
    chunk_wmma(ap1, bp1, acc);
    __syncthreads();       // buffer 1 free for refill
  }

  // D 16x16 f32 layout: VGPR v holds row (v + 8*half), column l16.
  const int bIdx = blockIdx.x * 4 + ctile;              // 0..127
  const int nrow = rowBase + rtile * 16 + 8 * half;     // sample index base
  float* dst = Mws + ((size_t)bIdx * N_SAMP + nrow) * CDIM + l16;
  #pragma unroll
  for (int v = 0; v < 8; ++v)
    dst[v * CDIM] = acc[v];
}

// ---------------------------------------------------------------------------
// Kernel 2: o[n,b] = sum_m exp(-sum_c |M[n,b,c]-M[m,b,c]|)
// One block per b; thread t = sample n. M[:,b,:] staged in 16KB LDS.
// ---------------------------------------------------------------------------
__global__ __launch_bounds__(256) void mbd_dist(
    const float* __restrict__ Mws,  // [128][256][16]
    float* __restrict__ out)        // [256, 8320]
{
  __shared__ float Mb[N_SAMP * CDIM];
  const int b = blockIdx.x;
  const int t = threadIdx.x;

  const float4* src = (const float4*)(Mws + (size_t)b * N_SAMP * CDIM);
  #pragma unroll
  for (int i = 0; i < 4; ++i)
    ((float4*)Mb)[i * 256 + t] = src[i * 256 + t];
  __syncthreads();

  float mine[CDIM];
  #pragma unroll
  for (int c = 0; c < CDIM; ++c) mine[c] = Mb[t * CDIM + c];

  float o = 0.f;
  for (int m = 0; m < N_SAMP; ++m) {
    const float* other = &Mb[m * CDIM];   // uniform address -> LDS broadcast
    float d = 0.f;
    #pragma unroll
    for (int c = 0; c < CDIM; ++c) d += fabsf(mine[c] - other[c]);
    o += __expf(-d);
  }
  out[(size_t)t * OUTW + FEAT + b] = o;
}

// ---------------------------------------------------------------------------
// Kernel 3: out[:, 0:8192] = inp (float4 copy into strided rows)
// ---------------------------------------------------------------------------
__global__ __launch_bounds__(256) void mbd_copy(
    const float* __restrict__ inp, float* __restrict__ out)
{
  const int i = blockIdx.x * 256 + threadIdx.x;  // float4 index, 524288 total
  const int row = i >> 11;                       // 2048 float4 per row
  const int col = i & 2047;
  const float4 v = ((const float4*)inp)[i];
  *(float4*)(out + (size_t)row * OUTW + col * 4) = v;
}

extern "C" void kernel_launch(void* const* d_in, const int* in_sizes, int n_in,
                              void* d_out, int out_size, void* d_ws, size_t ws_size,
                              hipStream_t stream) {
  (void)in_sizes; (void)n_in; (void)out_size; (void)ws_size;
  const float* inp = (const float*)d_in[0];   // [256, 8192]
  const float* T   = (const float*)d_in[1];   // [8192, 2048]
  float* out = (float*)d_out;                 // [256, 8320]
  float* Mws = (float*)d_ws;                  // [128][256][16] = 2 MB scratch

  mbd_gemm_wmma<<<dim3(32, 8), 256, 0, stream>>>(inp, T, Mws);
  mbd_dist<<<dim3(128), 256, 0, stream>>>(Mws, out);
  mbd_copy<<<dim3(2048), 256, 0, stream>>>(inp, out);
}